// HierarchicalSoftmax_26852135535128
// MI455X (gfx1250) — compile-verified
//
#include <hip/hip_runtime.h>

#define VOCAB     100000
#define N_NODES   (VOCAB - 1)
#define MAX_DEPTH 24
#define EMBED     128
#define BATCH     16384

typedef __attribute__((ext_vector_type(2))) float v2f;
typedef __attribute__((ext_vector_type(8))) float v8f;

// One wave32 per target id. D = A(16x4) x B(4x16) + C accumulated over 32 K-steps.
// A rows are all equal to the shared input vector chunk; B columns are 16 gathered
// rows of W (depth tile). Two tiles (d=0..15, d=16..23 + 8 masked pads) cover the
// padded Huffman path.
__global__ __launch_bounds__(128) void hsoftmax_wmma_kernel(
    const float* __restrict__ x,         // [EMBED]
    const int*   __restrict__ id_list,   // [BATCH]
    const float* __restrict__ W,         // [N_NODES, EMBED]
    const float* __restrict__ bias,      // [N_NODES]
    const int*   __restrict__ paths,     // [VOCAB, MAX_DEPTH]
    const int*   __restrict__ path_len,  // [VOCAB]
    float*       __restrict__ out)       // [BATCH]
{
    __shared__ float xs[EMBED];
    xs[threadIdx.x] = x[threadIdx.x];    // block = 128 threads, loads all of x
    __syncthreads();

    const int lane = threadIdx.x & 31;
    const int wid  = threadIdx.x >> 5;
    const int n    = blockIdx.x * 4 + wid;      // target index, grid covers BATCH exactly

    const int id  = id_list[n];
    const int len = path_len[id];

    const int d0   = lane & 15;   // column (depth within tile) this lane feeds/reads
    const int half = lane >> 4;   // selects K=0/1 vs K=2/3 of the 4-wide K chunk

    // Node indices for the two depth tiles (tile1 columns 8..15 are padding -> node 0,
    // masked later; row 0 is identical across lanes so its gather stays cache-hot)
    const int  nd0 = paths[id * MAX_DEPTH + d0];
    const bool t1v = (d0 < (MAX_DEPTH - 16));   // d0 < 8
    const int  nd1 = t1v ? paths[id * MAX_DEPTH + 16 + d0] : 0;

    const float* __restrict__ row0 = W + (size_t)nd0 * EMBED;
    const float* __restrict__ row1 = W + (size_t)nd1 * EMBED;

    v8f c0 = {};
    v8f c1 = {};

#pragma unroll 8
    for (int kc = 0; kc < EMBED / 4; ++kc) {
        const int k = kc * 4 + 2 * half;   // even -> 8-byte aligned float2 loads

        const float2 av = *reinterpret_cast<const float2*>(&xs[k]);
        const float2 w0 = *reinterpret_cast<const float2*>(row0 + k);
        const float2 w1 = *reinterpret_cast<const float2*>(row1 + k);

        v2f a;  a[0]  = av.x; a[1]  = av.y;   // A: every row = x[k0..k0+3]
        v2f b0; b0[0] = w0.x; b0[1] = w0.y;   // B: column d0 = W[nd0][k0..k0+3]
        v2f b1; b1[0] = w1.x; b1[1] = w1.y;

        // (neg_a, A, neg_b, B, c_mod, C, reuse_a, reuse_b)
        c0 = __builtin_amdgcn_wmma_f32_16x16x4_f32(false, a, false, b0,
                                                   (short)0, c0, false, false);
        c1 = __builtin_amdgcn_wmma_f32_16x16x4_f32(false, a, false, b1,
                                                   (short)0, c1, false, false);
    }

    // C/D layout: VGPR0, any lane -> M row (all identical), N = lane%16 = d0.
    const float l0 = c0[0] + bias[nd0];
    const float l1 = c1[0] + bias[nd1];

    // sigmoid(x) = rcp(1 + exp(-x)); v_rcp_f32 (~1 ulp) instead of IEEE divide.
    // Limits behave: exp(-x) -> inf => rcp(inf) = 0, matching sigmoid(-inf).
    const float e0 = __expf(-l0);
    const float e1 = __expf(-l1);
    float p0 = (d0 < len)               ? __builtin_amdgcn_rcpf(1.0f + e0) : 1.0f;
    float p1 = (t1v && (16 + d0) < len) ? __builtin_amdgcn_rcpf(1.0f + e1) : 1.0f;

    // Product over the 16 columns (lanes 0-15 and 16-31 hold duplicate copies).
    float v = p0 * p1;
    v *= __shfl_xor(v, 1, 32);
    v *= __shfl_xor(v, 2, 32);
    v *= __shfl_xor(v, 4, 32);
    v *= __shfl_xor(v, 8, 32);

    if (lane == 0) out[n] = v;
}

extern "C" void kernel_launch(void* const* d_in, const int* in_sizes, int n_in,
                              void* d_out, int out_size, void* d_ws, size_t ws_size,
                              hipStream_t stream) {
    const float* x        = (const float*)d_in[0];
    const int*   id_list  = (const int*)  d_in[1];
    const float* W        = (const float*)d_in[2];
    const float* bias     = (const float*)d_in[3];
    const int*   paths    = (const int*)  d_in[4];
    const int*   path_len = (const int*)  d_in[5];
    float*       out      = (float*)d_out;

    (void)in_sizes; (void)n_in; (void)out_size; (void)d_ws; (void)ws_size;

    dim3 grid(BATCH / 4);   // 4 waves (targets) per 128-thread block
    dim3 block(128);
    hipLaunchKernelGGL(hsoftmax_wmma_kernel, grid, block, 0, stream,
                       x, id_list, W, bias, paths, path_len, out);
}